// SparseGraphAttention_15479062135314
// MI455X (gfx1250) — compile-verified
//
#include <hip/hip_runtime.h>

// ---------------------------------------------------------------------------
// GAT layer for MI455X (gfx1250, wave32).
//   x[N,256] @ W[256,128] via V_WMMA_F32_16X16X4_F32 (exact fp32)
//   then edge-phase scatter softmax + scatter aggregation (L2-resident atomics)
// ---------------------------------------------------------------------------

typedef __attribute__((ext_vector_type(2))) float v2f;
typedef __attribute__((ext_vector_type(8))) float v8f;

#define IN_DIM 256
#define HD 128          // H*D = 4*32
#define NHEAD 4
#define NEG_SLOPE 0.2f

// ---- ordered-key helpers for float atomic max over signed values ----------
__device__ __forceinline__ unsigned fkey(float f) {
  unsigned u = __float_as_uint(f);
  return (u & 0x80000000u) ? ~u : (u | 0x80000000u);
}
__device__ __forceinline__ float funkey(unsigned k) {
  unsigned u = (k & 0x80000000u) ? (k & 0x7fffffffu) : ~k;
  return __uint_as_float(u);
}

// ---------------------------------------------------------------------------
// K0: zero/init accumulators (out, segmax bits, denom, global-max key)
// ---------------------------------------------------------------------------
__global__ void gat_init(float* __restrict__ out, unsigned* __restrict__ segmax,
                         float* __restrict__ denom, unsigned* __restrict__ gmax, int N) {
  int i = blockIdx.x * blockDim.x + threadIdx.x;
  if (i < N * HD) out[i] = 0.0f;
  if (i < N * NHEAD) { segmax[i] = 0u; denom[i] = 0.0f; }
  if (i == 0) *gmax = 0u;
}

// ---------------------------------------------------------------------------
// K1: x_proj = x @ W  via WMMA f32 16x16x4.  One wave per 16x16 output tile.
// A 16x4 layout : lane L holds M=L&15, K = k + 2*(L>=16) {+0,+1} in v0,v1
// B 4x16 layout : lane L holds N=L&15, K rows k+2*(L>=16) {+0,+1} in v0,v1
// C 16x16      : VGPR r, lane L -> M = r + 8*(L>=16), N = L&15
//
// Branchless inner loop: lanes whose row would be OOB load through a clamped
// row pointer instead; their garbage only reaches C rows M >= N, which are
// never stored. Store epilogue: wave-uniform full-tile fast path (N%16==0 for
// the real problem), predicated path only for a ragged last tile.
// ---------------------------------------------------------------------------
__global__ void gat_gemm_wmma(const float* __restrict__ x, const float* __restrict__ W,
                              float* __restrict__ xproj, int N) {
  int wave = (blockIdx.x * blockDim.x + threadIdx.x) >> 5;
  int lane = threadIdx.x & 31;
  const int nTiles = HD / 16;                 // 8
  int mTile = wave / nTiles;
  int nTile = wave - mTile * nTiles;
  int mBase = mTile * 16;
  if (mBase >= N) return;                     // wave-uniform
  int nBase = nTile * 16;

  int l15  = lane & 15;
  int kOff = (lane >> 4) << 1;                // 0 for lanes 0-15, 2 for 16-31
  int mRow = mBase + l15;
  int mClamped = (mRow < N) ? mRow : (N - 1);
  const float* xr = x + (size_t)mClamped * IN_DIM + kOff;
  const float* wc = W + (size_t)kOff * HD + nBase + l15;

  v8f c = {};
#pragma unroll 4
  for (int k = 0; k < IN_DIM; k += 4) {
    v2f a = *(const v2f*)(xr + k);            // global_load_b64 (K, K+1)
    v2f b;
    b.x = wc[(size_t)k * HD];                 // row K   of W
    b.y = wc[(size_t)k * HD + HD];            // row K+1 of W
    // (neg_a, A, neg_b, B, c_mod, C, reuse_a, reuse_b)
    c = __builtin_amdgcn_wmma_f32_16x16x4_f32(false, a, false, b, (short)0, c,
                                              false, false);
  }

  int mHalf = (lane >> 4) << 3;               // +0 or +8
  float* outBase = xproj + (size_t)(mBase + mHalf) * HD + nBase + l15;
  if (mBase + 16 <= N) {
    // full tile (wave-uniform): 8 unguarded stores
#pragma unroll
    for (int r = 0; r < 8; ++r) outBase[(size_t)r * HD] = c[r];
  } else {
    // ragged last tile: per-row predicated stores
#pragma unroll
    for (int r = 0; r < 8; ++r) {
      if (mBase + mHalf + r < N) outBase[(size_t)r * HD] = c[r];
    }
  }
}

// ---------------------------------------------------------------------------
// K2: al[n,h] = <x_proj[n,h,:], attn_l[h,:]>, likewise ar. One thread per (n,h).
// ---------------------------------------------------------------------------
__global__ void gat_node_attn(const float* __restrict__ xproj,
                              const float* __restrict__ attn_l,
                              const float* __restrict__ attn_r,
                              float* __restrict__ al, float* __restrict__ ar, int N) {
  int t = blockIdx.x * blockDim.x + threadIdx.x;
  if (t >= N * NHEAD) return;
  int n = t >> 2, h = t & 3;
  const float4* xp = (const float4*)(xproj + (size_t)n * HD + h * 32);
  const float4* wl = (const float4*)(attn_l + h * 32);
  const float4* wr = (const float4*)(attn_r + h * 32);
  float sl = 0.0f, sr = 0.0f;
#pragma unroll
  for (int i = 0; i < 8; ++i) {
    float4 v = xp[i], l = wl[i], r = wr[i];
    sl += v.x * l.x + v.y * l.y + v.z * l.z + v.w * l.w;
    sr += v.x * r.x + v.y * r.y + v.z * r.z + v.w * r.w;
  }
  al[t] = sl;
  ar[t] = sr;
}

// ---------------------------------------------------------------------------
// K3: alpha_pre = leaky_relu(al[row]+ar[col]); global max (wave-reduced key atomic)
// ---------------------------------------------------------------------------
__global__ void gat_edge_alpha(const long long* __restrict__ ei,
                               const float* __restrict__ al, const float* __restrict__ ar,
                               float* __restrict__ alpha, unsigned* __restrict__ gmax,
                               int E) {
  int e = blockIdx.x * blockDim.x + threadIdx.x;
  float m = -3.4e38f;
  if (e < E) {
    int row = (int)ei[e];
    int col = (int)ei[(size_t)E + e];
#pragma unroll
    for (int h = 0; h < NHEAD; ++h) {
      float a = al[(size_t)row * NHEAD + h] + ar[(size_t)col * NHEAD + h];
      a = (a >= 0.0f) ? a : NEG_SLOPE * a;
      alpha[(size_t)e * NHEAD + h] = a;
      m = fmaxf(m, a);
    }
  }
#pragma unroll
  for (int off = 16; off > 0; off >>= 1) m = fmaxf(m, __shfl_xor(m, off, 32));
  if ((threadIdx.x & 31) == 0) atomicMax(gmax, fkey(m));
}

// ---------------------------------------------------------------------------
// K4: alpha = exp(alpha - gmax); segment max per destination node (bits atomicMax,
//     valid since exp >= 0 -> raw float bits are monotone)
// ---------------------------------------------------------------------------
__global__ void gat_edge_segmax(const long long* __restrict__ ei,
                                float* __restrict__ alpha,
                                const unsigned* __restrict__ gmax,
                                unsigned* __restrict__ segmax, int E) {
  int e = blockIdx.x * blockDim.x + threadIdx.x;
  if (e >= E) return;
  float g = funkey(*gmax);
  int row = (int)ei[e];
#pragma unroll
  for (int h = 0; h < NHEAD; ++h) {
    float a = expf(alpha[(size_t)e * NHEAD + h] - g);
    alpha[(size_t)e * NHEAD + h] = a;
    atomicMax(&segmax[(size_t)row * NHEAD + h], __float_as_uint(a));
  }
}

// ---------------------------------------------------------------------------
// K5: a = exp(alpha - segmax[row]); denom[row] += a
// ---------------------------------------------------------------------------
__global__ void gat_edge_denom(const long long* __restrict__ ei,
                               float* __restrict__ alpha,
                               const unsigned* __restrict__ segmax,
                               float* __restrict__ denom, int E) {
  int e = blockIdx.x * blockDim.x + threadIdx.x;
  if (e >= E) return;
  int row = (int)ei[e];
#pragma unroll
  for (int h = 0; h < NHEAD; ++h) {
    float sm = __uint_as_float(segmax[(size_t)row * NHEAD + h]);
    float a = expf(alpha[(size_t)e * NHEAD + h] - sm);
    alpha[(size_t)e * NHEAD + h] = a;
    atomicAdd(&denom[(size_t)row * NHEAD + h], a);
  }
}

// ---------------------------------------------------------------------------
// K6: out[row] += (a/denom[row]) * x_proj[col]. One wave per edge; lane owns 4
//     contiguous floats (float4 gather, 4 atomic f32 adds -> L2 resident).
// ---------------------------------------------------------------------------
__global__ void gat_edge_aggregate(const long long* __restrict__ ei,
                                   const float* __restrict__ alpha,
                                   const float* __restrict__ denom,
                                   const float* __restrict__ xproj,
                                   float* __restrict__ out, int E) {
  int wave = (int)(((size_t)blockIdx.x * blockDim.x + threadIdx.x) >> 5);
  int lane = threadIdx.x & 31;
  if (wave >= E) return;
  int row = (int)ei[wave];
  int col = (int)ei[(size_t)E + wave];
  int h = lane >> 3;                                 // 32 floats per head / 4 per lane
  float coeff = alpha[(size_t)wave * NHEAD + h] / denom[(size_t)row * NHEAD + h];
  float4 v = ((const float4*)(xproj + (size_t)col * HD))[lane];
  float* o = out + (size_t)row * HD + lane * 4;
  atomicAdd(o + 0, coeff * v.x);
  atomicAdd(o + 1, coeff * v.y);
  atomicAdd(o + 2, coeff * v.z);
  atomicAdd(o + 3, coeff * v.w);
}

// ---------------------------------------------------------------------------
extern "C" void kernel_launch(void* const* d_in, const int* in_sizes, int n_in,
                              void* d_out, int out_size, void* d_ws, size_t ws_size,
                              hipStream_t stream) {
  const float* x          = (const float*)d_in[0];
  const long long* ei     = (const long long*)d_in[1];
  const float* W          = (const float*)d_in[2];
  const float* attn_l     = (const float*)d_in[3];
  const float* attn_r     = (const float*)d_in[4];
  float* out              = (float*)d_out;

  int N = in_sizes[0] / IN_DIM;      // 50000
  int E = in_sizes[1] / 2;           // 1600000

  // workspace layout (all 4B-aligned; xproj at 0 keeps float4 16B alignment)
  char* p = (char*)d_ws;
  float* xproj   = (float*)p;  p += (size_t)N * HD * sizeof(float);
  float* al      = (float*)p;  p += (size_t)N * NHEAD * sizeof(float);
  float* ar      = (float*)p;  p += (size_t)N * NHEAD * sizeof(float);
  unsigned* smax = (unsigned*)p; p += (size_t)N * NHEAD * sizeof(unsigned);
  float* denom   = (float*)p;  p += (size_t)N * NHEAD * sizeof(float);
  float* alpha   = (float*)p;  p += (size_t)E * NHEAD * sizeof(float);
  unsigned* gmax = (unsigned*)p;

  const int T = 256;

  gat_init<<<(N * HD + T - 1) / T, T, 0, stream>>>(out, smax, denom, gmax, N);

  int mTiles = (N + 15) / 16;        // 8 waves/block == 8 N-tiles for one M-tile
  gat_gemm_wmma<<<mTiles, T, 0, stream>>>(x, W, xproj, N);

  gat_node_attn<<<(N * NHEAD + T - 1) / T, T, 0, stream>>>(xproj, attn_l, attn_r,
                                                           al, ar, N);

  gat_edge_alpha<<<(E + T - 1) / T, T, 0, stream>>>(ei, al, ar, alpha, gmax, E);
  gat_edge_segmax<<<(E + T - 1) / T, T, 0, stream>>>(ei, alpha, gmax, smax, E);
  gat_edge_denom<<<(E + T - 1) / T, T, 0, stream>>>(ei, alpha, smax, denom, E);

  size_t aggThreads = (size_t)E * 32;
  gat_edge_aggregate<<<(unsigned)((aggThreads + T - 1) / T), T, 0, stream>>>(
      ei, alpha, denom, xproj, out, E);
}